// Linear_act_sp_8847632630469
// MI455X (gfx1250) — compile-verified
//
#include <hip/hip_runtime.h>
#include <hip/hip_bf16.h>

// ---------------------------------------------------------------------------
// Linear with activation smoothing + 2:4 semi-structured prune, on MI455X.
//   s[k] = sqrt(colmax|x| / max(colmax|W|, 1e-8))
//   out  = prune_2of4(x/s) @ (W*s)^T        x:[8192,4096], W:[4096,4096]
//
// Compute-bound (275 GFLOP vs ~335 MB HBM). The prune is exactly CDNA5's
// 2:4 SWMMAC sparsity -> V_SWMMAC_F32_16X16X64_BF16.
// This round removes the staging redundancy: precompute bf16(W*s) and the
// compressed 2:4 representation of x/s once into d_ws (~68 MB), then run a
// GEMM whose tiles are DMA'd into LDS by the Tensor Data Mover
// (TENSOR_LOAD_TO_LDS + S_WAIT_TENSORCNT, double-buffered), so the K-loop is
// pure ds_load + v_swmmac. Falls back to the round-2 fused kernel when
// ws_size is too small or the TDM builtin is unavailable.
// ---------------------------------------------------------------------------

typedef __attribute__((ext_vector_type(32))) __bf16 v32bf;
typedef __attribute__((ext_vector_type(16))) __bf16 v16bf;
typedef __attribute__((ext_vector_type(8)))  __bf16 v8bf;
typedef __attribute__((ext_vector_type(4)))  __bf16 v4bf;
typedef __attribute__((ext_vector_type(8)))  float  v8f;
typedef __attribute__((ext_vector_type(4)))  unsigned int u32x4;
typedef __attribute__((ext_vector_type(8)))  int i32x8;
typedef __attribute__((ext_vector_type(4)))  int i32x4;

#define D_IN   4096
#define D_OUT  4096
#define NROWS  8192        // 4 * 2048
#define BM     128
#define BN     128
#define SBK    64          // expanded K per swmmac tile
#define KC     (D_IN / 2)  // compressed row length of Xc (2048)
#define NIW    (D_IN / 32) // index words per row of XI (128)

#if defined(__has_builtin)
#if __has_builtin(__builtin_amdgcn_tensor_load_to_lds)
#define HAVE_TDM 1
#endif
#endif

// ---------------- scale pass ------------------------------------------------

__global__ void asp_init_kernel(float* p, int n) {
    int i = blockIdx.x * blockDim.x + threadIdx.x;
    if (i < n) p[i] = 0.0f;
}

// column-wise max(|X|): block-partial reduction + atomicMax on uint bits
// (bit compare == float compare for non-negative floats; order independent)
__global__ void asp_colmax_kernel(const float* __restrict__ X,
                                  float* __restrict__ out,
                                  int rows, int cols) {
    int c   = blockIdx.x * blockDim.x + threadIdx.x;
    int per = rows / gridDim.y;
    int r0  = blockIdx.y * per;
    float m = 0.0f;
    for (int r = r0; r < r0 + per; ++r)
        m = fmaxf(m, fabsf(X[(size_t)r * cols + c]));
    atomicMax((unsigned int*)(out + c), __float_as_uint(m));
}

__global__ void asp_scale_kernel(const float* __restrict__ ma,
                                 const float* __restrict__ mw,
                                 float* __restrict__ s,
                                 float* __restrict__ rs, int n) {
    int i = blockIdx.x * blockDim.x + threadIdx.x;
    if (i < n) {
        float sv = sqrtf(ma[i] / fmaxf(mw[i], 1e-8f));
        s[i]  = sv;
        rs[i] = 1.0f / sv;   // one IEEE divide per column, outside the GEMM
    }
}

// 2:4 prune of a group of 4: keep top-2 by |.|, lowest index wins ties
// (matches lax.top_k). Outputs kept values k0,k1 (index order), positions i0<i1.
__device__ __forceinline__ void prune24(float v0, float v1, float v2, float v3,
                                        float& k0, float& k1, int& i0, int& i1) {
    const float a0 = fabsf(v0), a1 = fabsf(v1), a2 = fabsf(v2), a3 = fabsf(v3);
    const int r0 = (a1 >  a0) + (a2 >  a0) + (a3 >  a0);
    const int r1 = (a0 >= a1) + (a2 >  a1) + (a3 >  a1);
    const int r2 = (a0 >= a2) + (a1 >= a2) + (a3 >  a2);
    const int r3 = (a0 >= a3) + (a1 >= a3) + (a2 >= a3);
    const bool f0 = r0 < 2, f1 = r1 < 2, f2 = r2 < 2, f3 = r3 < 2;
    i0 = f0 ? 0 : (f1 ? 1 : 2);
    i1 = f3 ? 3 : (f2 ? 2 : 1);
    k0 = f0 ? v0 : (f1 ? v1 : v2);
    k1 = f3 ? v3 : (f2 ? v2 : v1);
}

// ---------------- precompute: compressed X and scaled bf16 W ----------------

// one thread per (row, 32-expanded-K chunk): 16 kept bf16 + one 2-bit-pair word
__global__ __launch_bounds__(256)
void asp_compress_x_kernel(const float* __restrict__ X,
                           const float* __restrict__ RS,
                           __bf16* __restrict__ Xc,
                           unsigned int* __restrict__ XI) {
    const int idx = blockIdx.x * 256 + threadIdx.x;   // NROWS * NIW threads
    const int row = idx >> 7;                         // / NIW
    const int ch  = idx & (NIW - 1);
    const float* xp = X  + (size_t)row * D_IN + ch * 32;
    const float* rp = RS + ch * 32;
    unsigned int idxw = 0;
    v16bf av;
#pragma unroll
    for (int g = 0; g < 8; ++g) {
        const float4 xv = *(const float4*)(xp + g * 4);
        const float4 rv = *(const float4*)(rp + g * 4);
        float k0, k1; int i0, i1;
        prune24(xv.x * rv.x, xv.y * rv.y, xv.z * rv.z, xv.w * rv.w, k0, k1, i0, i1);
        av[g * 2 + 0] = (__bf16)k0;
        av[g * 2 + 1] = (__bf16)k1;
        idxw |= (unsigned int)(i0 | (i1 << 2)) << (4 * g);
    }
    *(v16bf*)(Xc + (size_t)row * KC + ch * 16) = av;
    XI[(size_t)row * NIW + ch] = idxw;
}

// Wbf[n][k] = bf16(W[n][k] * s[k]); one float4 group per thread
__global__ __launch_bounds__(256)
void asp_scale_w_kernel(const float* __restrict__ W,
                        const float* __restrict__ S,
                        __bf16* __restrict__ Wbf) {
    const int idx = blockIdx.x * 256 + threadIdx.x;   // D_OUT * D_IN / 4 threads
    const int row = idx >> 10;
    const int kg  = idx & 1023;
    const float4 wv = *(const float4*)(W + (size_t)row * D_IN + kg * 4);
    const float4 sv = *(const float4*)(S + kg * 4);
    v4bf wk;
    wk[0] = (__bf16)(wv.x * sv.x); wk[1] = (__bf16)(wv.y * sv.y);
    wk[2] = (__bf16)(wv.z * sv.z); wk[3] = (__bf16)(wv.w * sv.w);
    *(v4bf*)(Wbf + (size_t)row * D_IN + kg * 4) = wk;
}

// ================== TDM-fed double-buffered SWMMAC GEMM =====================
#if HAVE_TDM

#if __has_include(<hip/amd_detail/amd_gfx1250_TDM.h>)
// therock toolchain: 6-arg builtin (extra int32x8 group)
#define TDM_CALL(g0, g1, g2, g3) \
    __builtin_amdgcn_tensor_load_to_lds(g0, g1, g2, g3, (i32x8)0, 0)
#else
// ROCm 7.2: 5-arg builtin
#define TDM_CALL(g0, g1, g2, g3) \
    __builtin_amdgcn_tensor_load_to_lds(g0, g1, g2, g3, 0)
#endif

#if __has_builtin(__builtin_amdgcn_s_wait_tensorcnt)
#define WAIT_TENSORCNT(n) __builtin_amdgcn_s_wait_tensorcnt(n)
#else
#define WAIT_TENSORCNT(n) asm volatile("s_wait_tensorcnt " #n ::: "memory")
#endif

// Build a 2D tensor-DMA descriptor (D# groups, ISA ch.8) and issue the load.
// dims/strides in data_size units; dsz_log2: 0=1B,1=2B,2=4B.
__device__ __forceinline__ void tdm_load_2d(unsigned int lds_off, const void* gptr,
                                            unsigned int tensor_d0, unsigned int tensor_d1,
                                            unsigned int tile_d0, unsigned int tile_d1,
                                            unsigned int stride0, unsigned int dsz_log2) {
    const unsigned long long ga = (unsigned long long)gptr;
    u32x4 g0;
    g0[0] = 1u;                                            // count=1, user mode
    g0[1] = lds_off;                                       // LDS byte address
    g0[2] = (unsigned int)(ga & 0xFFFFFFFFu);              // global_addr[31:0]
    g0[3] = (unsigned int)((ga >> 32) & 0x01FFFFFFu)       // global_addr[56:32]
          | (2u << 30);                                    // type = 2 ("image")
    i32x8 g1;
    g1[0] = (int)(dsz_log2 << 16);                         // data_size
    g1[1] = (int)((tensor_d0 & 0xFFFFu) << 16);            // tensor_dim0[15:0]
    g1[2] = (int)(((tensor_d0 >> 16) & 0xFFFFu)            // tensor_dim0[31:16]
          | ((tensor_d1 & 0xFFFFu) << 16));                // tensor_dim1[15:0]
    g1[3] = (int)(((tensor_d1 >> 16) & 0xFFFFu)            // tensor_dim1[31:16]
          | (tile_d0 << 16));                              // tile_dim0
    g1[4] = (int)tile_d1;                                  // tile_dim1 (tile_dim2=0)
    g1[5] = (int)stride0;                                  // tensor_dim0_stride lo
    g1[6] = 0;                                             // stride0 hi / stride1 lo
    g1[7] = 0;
    const i32x4 z = {0, 0, 0, 0};                          // 2D: groups 2/3 unused
    TDM_CALL(g0, g1, z, z);
}

__global__ __launch_bounds__(256)
void asp_swmmac_tdm_kernel(const __bf16* __restrict__ Xc,
                           const unsigned int* __restrict__ XI,
                           const __bf16* __restrict__ Wbf,
                           float* __restrict__ out) {
    __shared__ __attribute__((aligned(128))) __bf16       Ac[2][BM][32];
    __shared__ __attribute__((aligned(128))) unsigned int AI[2][BM][2];
    __shared__ __attribute__((aligned(128))) __bf16       Bs[2][BN][SBK];

    const int tid  = threadIdx.x;
    const int lane = tid & 31;
    const int wid  = tid >> 5;        // 8 waves
    const int wy   = wid >> 2;        // 0..1  -> 64 rows of M
    const int wx   = wid & 3;         // 0..3  -> 32 cols of N
    const int hl   = lane >> 4;
    const int l15  = lane & 15;
    const int m0   = blockIdx.y * BM;
    const int n0   = blockIdx.x * BN;

    v8f acc[4][2] = {};

    // issue the 3 DMAs for one K-tile into LDS buffer b (wave 0 only)
    auto issue_tile = [&](int b, int kt) {
        // compressed A values: 128 rows x 32 bf16
        tdm_load_2d((unsigned int)(size_t)&Ac[b][0][0],
                    Xc + (size_t)m0 * KC + kt / 2,
                    KC, NROWS, 32, BM, KC, /*2B*/1);
        // 2:4 index words: 128 rows x 2 u32 (each covers 32 expanded K)
        tdm_load_2d((unsigned int)(size_t)&AI[b][0][0],
                    XI + (size_t)m0 * NIW + kt / 32,
                    NIW, NROWS, 2, BM, NIW, /*4B*/2);
        // dense B: 128 rows x 64 bf16
        tdm_load_2d((unsigned int)(size_t)&Bs[b][0][0],
                    Wbf + (size_t)n0 * D_IN + kt,
                    D_IN, D_OUT, SBK, BN, D_IN, /*2B*/1);
    };

    if (wid == 0) issue_tile(0, 0);

    const int ntiles = D_IN / SBK;
    for (int t = 0; t < ntiles; ++t) {
        const int cur = t & 1;
        if (wid == 0) {
            if (t + 1 < ntiles) {
                issue_tile(cur ^ 1, (t + 1) * SBK);  // prefetch next tile
                WAIT_TENSORCNT(3);                   // current tile's 3 done
            } else {
                WAIT_TENSORCNT(0);
            }
        }
        __syncthreads();                             // publish LDS to all waves

        // B fragments (dense 64x16): lane = col l15, two 16-element runs
        v32bf bfrag[2];
#pragma unroll
        for (int j = 0; j < 2; ++j) {
            const int n = wx * 32 + j * 16 + l15;
            const v16bf blo = *(const v16bf*)&Bs[cur][n][hl * 16];
            const v16bf bhi = *(const v16bf*)&Bs[cur][n][32 + hl * 16];
#pragma unroll
            for (int e = 0; e < 16; ++e) { bfrag[j][e] = blo[e]; bfrag[j][16 + e] = bhi[e]; }
        }
#pragma unroll
        for (int i = 0; i < 4; ++i) {
            const int m = wy * 64 + i * 16 + l15;
            const v8bf alo = *(const v8bf*)&Ac[cur][m][hl * 8];
            const v8bf ahi = *(const v8bf*)&Ac[cur][m][hl * 8 + 16];
            v16bf af;
#pragma unroll
            for (int e = 0; e < 8; ++e) { af[e] = alo[e]; af[e + 8] = ahi[e]; }
            const int aidx = (int)AI[cur][m][hl];
#pragma unroll
            for (int j = 0; j < 2; ++j)
                acc[i][j] = __builtin_amdgcn_swmmac_f32_16x16x64_bf16(
                    false, af, false, bfrag[j], acc[i][j], aidx, false, false);
        }
        __syncthreads();   // done reading buffer before TDM reuses it
    }

#pragma unroll
    for (int i = 0; i < 4; ++i) {
        const int mrow = m0 + wy * 64 + i * 16 + hl * 8;
#pragma unroll
        for (int j = 0; j < 2; ++j) {
            const int ncol = n0 + wx * 32 + j * 16 + l15;
            float* op = out + (size_t)mrow * D_OUT + ncol;
#pragma unroll
            for (int r = 0; r < 8; ++r)
                op[(size_t)r * D_OUT] = acc[i][j][r];
        }
    }
}
#endif  // HAVE_TDM

// ================== fused fallback (round-2) ======================

__global__ __launch_bounds__(256)
void asp_swmmac_fused_kernel(const float* __restrict__ X,
                             const float* __restrict__ W,
                             const float* __restrict__ S,
                             const float* __restrict__ RS,
                             float* __restrict__ out) {
    __shared__ __bf16       Ac[BM][32];
    __shared__ unsigned int AI[BM][2];
    __shared__ __bf16       Bs[BN][SBK];

    const int tid  = threadIdx.x;
    const int lane = tid & 31;
    const int wid  = tid >> 5;
    const int wy   = wid >> 2;
    const int wx   = wid & 3;
    const int hl   = lane >> 4;
    const int l15  = lane & 15;
    const int m0   = blockIdx.y * BM;
    const int n0   = blockIdx.x * BN;
    const int arow  = tid >> 1;
    const int ahalf = tid & 1;

    v8f acc[4][2] = {};

    for (int kt = 0; kt < D_IN; kt += SBK) {
        {
            const float* xp = X  + (size_t)(m0 + arow) * D_IN + kt + ahalf * 32;
            const float* rp = RS + kt + ahalf * 32;
            unsigned int idxw = 0;
            v16bf av;
#pragma unroll
            for (int g = 0; g < 8; ++g) {
                const float4 xv = *(const float4*)(xp + g * 4);
                const float4 rv = *(const float4*)(rp + g * 4);
                float k0, k1; int i0, i1;
                prune24(xv.x * rv.x, xv.y * rv.y, xv.z * rv.z, xv.w * rv.w,
                        k0, k1, i0, i1);
                av[g * 2 + 0] = (__bf16)k0;
                av[g * 2 + 1] = (__bf16)k1;
                idxw |= (unsigned int)(i0 | (i1 << 2)) << (4 * g);
            }
            *(v16bf*)&Ac[arow][ahalf * 16] = av;
            AI[arow][ahalf] = idxw;
            if (kt + SBK < D_IN) __builtin_prefetch(xp + SBK, 0, 3);
        }
#pragma unroll
        for (int i = 0; i < 8; ++i) {
            const int idx = tid + i * 256;
            const int row = idx >> 4;
            const int kg  = idx & 15;
            const float* wp = W + (size_t)(n0 + row) * D_IN + kt + kg * 4;
            const float4 wv = *(const float4*)wp;
            const float4 sv = *(const float4*)(S + kt + kg * 4);
            v4bf wk;
            wk[0] = (__bf16)(wv.x * sv.x); wk[1] = (__bf16)(wv.y * sv.y);
            wk[2] = (__bf16)(wv.z * sv.z); wk[3] = (__bf16)(wv.w * sv.w);
            *(v4bf*)&Bs[row][kg * 4] = wk;
            if (kt + SBK < D_IN) __builtin_prefetch(wp + SBK, 0, 3);
        }
        __syncthreads();

        v32bf bfrag[2];
#pragma unroll
        for (int j = 0; j < 2; ++j) {
            const int n = wx * 32 + j * 16 + l15;
            const v16bf blo = *(const v16bf*)&Bs[n][hl * 16];
            const v16bf bhi = *(const v16bf*)&Bs[n][32 + hl * 16];
#pragma unroll
            for (int e = 0; e < 16; ++e) { bfrag[j][e] = blo[e]; bfrag[j][16 + e] = bhi[e]; }
        }
#pragma unroll
        for (int i = 0; i < 4; ++i) {
            const int m = wy * 64 + i * 16 + l15;
            const v8bf alo = *(const v8bf*)&Ac[m][hl * 8];
            const v8bf ahi = *(const v8bf*)&Ac[m][hl * 8 + 16];
            v16bf af;
#pragma unroll
            for (int e = 0; e < 8; ++e) { af[e] = alo[e]; af[e + 8] = ahi[e]; }
            const int aidx = (int)AI[m][hl];
#pragma unroll
            for (int j = 0; j < 2; ++j)
                acc[i][j] = __builtin_amdgcn_swmmac_f32_16x16x64_bf16(
                    false, af, false, bfrag[j], acc[i][j], aidx, false, false);
        }
        __syncthreads();
    }

#pragma unroll
    for (int i = 0; i < 4; ++i) {
        const int mrow = m0 + wy * 64 + i * 16 + hl * 8;
#pragma unroll
        for (int j = 0; j < 2; ++j) {
            const int ncol = n0 + wx * 32 + j * 16 + l15;
            float* op = out + (size_t)mrow * D_OUT + ncol;
#pragma unroll
            for (int r = 0; r < 8; ++r)
                op[(size_t)r * D_OUT] = acc[i][j][r];
        }
    }
}

// ---------------- launch ----------------------------------------------------

extern "C" void kernel_launch(void* const* d_in, const int* in_sizes, int n_in,
                              void* d_out, int out_size, void* d_ws, size_t ws_size,
                              hipStream_t stream) {
    const float* X = (const float*)d_in[0];   // [8192, 4096]
    const float* W = (const float*)d_in[1];   // [4096, 4096] (out, in)
    float* out = (float*)d_out;

    float* max_act = (float*)d_ws;            // [4096]
    float* max_w   = max_act + D_IN;          // [4096]
    float* s       = max_w + D_IN;            // [4096]
    float* rs      = s + D_IN;                // [4096]

    asp_init_kernel<<<dim3((2 * D_IN + 255) / 256), 256, 0, stream>>>(max_act, 2 * D_IN);
    asp_colmax_kernel<<<dim3(D_IN / 256, 32), 256, 0, stream>>>(X, max_act, NROWS, D_IN);
    asp_colmax_kernel<<<dim3(D_IN / 256, 16), 256, 0, stream>>>(W, max_w, D_OUT, D_IN);
    asp_scale_kernel<<<dim3(D_IN / 256), 256, 0, stream>>>(max_act, max_w, s, rs, D_IN);

    const dim3 gemm_grid(D_OUT / BN, NROWS / BM);

#if HAVE_TDM
    // scratch layout: 64KB scales | Xc 32MB | XI 4MB | Wbf 32MB
    char*           base = (char*)d_ws + 4 * (size_t)D_IN * sizeof(float);
    __bf16*         Xc   = (__bf16*)base;
    unsigned int*   XI   = (unsigned int*)(base + (size_t)NROWS * KC * sizeof(__bf16));
    __bf16*         Wbf  = (__bf16*)((char*)XI + (size_t)NROWS * NIW * sizeof(unsigned int));
    const size_t need = ((char*)(Wbf + (size_t)D_OUT * D_IN)) - (char*)d_ws;

    if (ws_size >= need) {
        asp_compress_x_kernel<<<dim3((size_t)NROWS * NIW / 256), 256, 0, stream>>>(X, rs, Xc, XI);
        asp_scale_w_kernel<<<dim3((size_t)D_OUT * D_IN / 4 / 256), 256, 0, stream>>>(W, s, Wbf);
        asp_swmmac_tdm_kernel<<<gemm_grid, 256, 0, stream>>>(Xc, XI, Wbf, out);
        return;
    }
#endif
    asp_swmmac_fused_kernel<<<gemm_grid, 256, 0, stream>>>(X, W, s, rs, out);
}